// SSIMLoss_64682207478335
// MI455X (gfx1250) — compile-verified
//
#include <hip/hip_runtime.h>

// ---------------------------------------------------------------------------
// SSIM loss, fused single-pass kernel for MI455X (gfx1250, wave32, WMMA).
//   img1, img2 : [32, 3, 512, 512] f32
//   out        : scalar f32 = 1 - mean(ssim_map)
// Per block: one 64x64 output tile of one (b,c) plane.
//   Phase A: load 74x74 zero-padded raw tiles (img1,img2) -> LDS
//   Phase B: horizontal 11-tap Gaussian on 5 maps (x1,x2,x1^2,x2^2,x1*x2)
//            as Toeplitz matmul: D = A(data 16x28) x B(weights 28x16)
//            on V_WMMA_F32_16X16X4_F32 -> LDS
//   Phase C: vertical 11-tap pass as D = A(weights 16x28) x B(data 28x16),
//            then elementwise SSIM on the v8f accumulators
//   Phase D: deterministic two-stage reduction via d_ws
// Both conv passes run on the matrix pipe; VALU is left with address math,
// the 3 product muls per K-chunk and the SSIM epilogue -> kernel is HBM-bound.
// ---------------------------------------------------------------------------

typedef float v2f __attribute__((ext_vector_type(2)));
typedef float v8f __attribute__((ext_vector_type(8)));

constexpr int IMGH   = 512;
constexpr int IMGW   = 512;
constexpr int TILE   = 64;           // output tile edge
constexpr int RROWS  = TILE + 10;    // 74 region rows/cols (halo = 5 each side)
constexpr int RS     = 77;           // raw LDS row stride
constexpr int RSZ    = RROWS * RS;   // 5698 floats per raw map
constexpr int HS     = 76;           // hconv stride: 8*76%64=32, 2*76%64=24 ->
                                     // half-wave bank-disjoint stores AND loads
constexpr int HSZ    = RROWS * HS;   // 5624 floats per hconv map
constexpr int OFF_RAWB = RSZ;              // 5698
constexpr int OFF_H    = 2 * RSZ;          // 11396
constexpr int OFF_RED  = OFF_H + 5 * HSZ;  // 39516
constexpr int LDS_FLOATS = OFF_RED + 8;    // 39524 -> 158096 bytes (2 WG/WGP)
constexpr int NBLOCKS  = 32 * 3 * (IMGH / TILE) * (IMGW / TILE); // 6144
constexpr float C1c = 0.01f * 0.01f;
constexpr float C2c = 0.03f * 0.03f;

// Gaussian(sigma=1.5, K=11), normalized; computed in double, matches reference.
__device__ constexpr float KW[11] = {
    0.00102838f, 0.00759876f, 0.03600077f, 0.10936069f, 0.21300554f,
    0.26601173f,
    0.21300554f, 0.10936069f, 0.03600077f, 0.00759876f, 0.00102838f};

// Toeplitz weight lookup: w[d] if 0<=d<=10 else 0 (register-only selects).
__device__ __forceinline__ float wsel(int d) {
    float r = 0.0f;
#pragma unroll
    for (int t = 0; t < 11; ++t) r = (d == t) ? KW[t] : r;
    return r;
}

#if __has_builtin(__builtin_amdgcn_wmma_f32_16x16x4_f32)
#define SSIM_HAVE_WMMA 1
#endif

__global__ __launch_bounds__(256) void ssim_fused_kernel(
    const float* __restrict__ img1, const float* __restrict__ img2,
    float* __restrict__ ws) {
    extern __shared__ float lds[];

    const int tid   = threadIdx.x;
    const int plane = blockIdx.x / ((IMGH / TILE) * (IMGW / TILE)); // 0..95
    const int tile  = blockIdx.x % ((IMGH / TILE) * (IMGW / TILE)); // 0..63
    const int tr    = tile >> 3;
    const int tc    = tile & 7;
    const int base_r = tr * TILE - 5;
    const int base_c = tc * TILE - 5;
    const size_t planeBase = (size_t)plane * (IMGH * IMGW);

    // ---------------- Phase A: raw tiles (zero-padded) -> LDS --------------
    for (int i = tid; i < RROWS * RROWS; i += 256) {
        const int rr = i / RROWS;
        const int cc = i - rr * RROWS;
        const int gr = base_r + rr;
        const int gc = base_c + cc;
        float va = 0.0f, vb = 0.0f;
        if (gr >= 0 && gr < IMGH && gc >= 0 && gc < IMGW) {
            const size_t g = planeBase + (size_t)gr * IMGW + gc;
            va = img1[g];
            vb = img2[g];
        }
        lds[rr * RS + cc]            = va;
        lds[OFF_RAWB + rr * RS + cc] = vb;
    }
    __syncthreads();

    float lsum = 0.0f;
    const int lane = tid & 31;
    const int wave = tid >> 5;

#if defined(SSIM_HAVE_WMMA)
    const int n  = lane & 15;  // matrix row (A) / column (B,D) held by lane
    const int hh = lane >> 4;  // half-wave selects K slots {0,1}+4j / {2,3}+4j

    // Gaussian-Toeplitz weight chunks w[s - n]: serve as the B operand of the
    // horizontal pass AND the A operand of the vertical pass.
    float wwx[7], wwy[7];
#pragma unroll
    for (int j = 0; j < 7; ++j) {
        const int s0 = 4 * j + (hh ? 2 : 0);
        wwx[j] = wsel(s0 - n);
        wwy[j] = wsel(s0 + 1 - n);
    }

    // ------------- Phase B: horizontal pass on WMMA, 5 maps -> LDS ---------
    // D(16 region-rows x 16 out-cols) = A(raw data 16x28) x B(weights 28x16).
    // 20 subtiles: 5 row-blocks (80 rows, rows >=74 clamped+discarded) x 4.
    for (int sub = wave; sub < 20; sub += 8) {  // uniform per wave
        const int R0h = (sub >> 2) * 16;
        const int c0h = (sub & 3) * 16;
        int rowm = R0h + n; if (rowm > RROWS - 1) rowm = RROWS - 1;
        const int rbase = rowm * RS;
        v8f dacc[5];
#pragma unroll
        for (int m5 = 0; m5 < 5; ++m5)
            dacc[m5] = (v8f){0.f, 0.f, 0.f, 0.f, 0.f, 0.f, 0.f, 0.f};
#pragma unroll
        for (int j = 0; j < 7; ++j) {
            int cs0 = c0h + 4 * j + (hh ? 2 : 0);
            int cs1 = cs0 + 1;
            if (cs0 > RROWS - 1) cs0 = RROWS - 1;  // zero-weight K columns
            if (cs1 > RROWS - 1) cs1 = RROWS - 1;
            const float x0 = lds[rbase + cs0];
            const float x1 = lds[rbase + cs1];
            const float y0 = lds[OFF_RAWB + rbase + cs0];
            const float y1 = lds[OFF_RAWB + rbase + cs1];
            v2f bw; bw.x = wwx[j]; bw.y = wwy[j];
            v2f pa;  pa.x = x0;      pa.y = x1;
            v2f pb;  pb.x = y0;      pb.y = y1;
            const v2f paa = pa * pa;
            const v2f pbb = pb * pb;
            const v2f pab = pa * pb;
            dacc[0] = __builtin_amdgcn_wmma_f32_16x16x4_f32(
                false, pa,  false, bw, (short)0, dacc[0], false, false);
            dacc[1] = __builtin_amdgcn_wmma_f32_16x16x4_f32(
                false, pb,  false, bw, (short)0, dacc[1], false, false);
            dacc[2] = __builtin_amdgcn_wmma_f32_16x16x4_f32(
                false, paa, false, bw, (short)0, dacc[2], false, false);
            dacc[3] = __builtin_amdgcn_wmma_f32_16x16x4_f32(
                false, pbb, false, bw, (short)0, dacc[3], false, false);
            dacc[4] = __builtin_amdgcn_wmma_f32_16x16x4_f32(
                false, pab, false, bw, (short)0, dacc[4], false, false);
        }
        // D layout: VGPR e holds region row R0h+e (lanes 0-15) / R0h+e+8
        // (lanes 16-31), column c0h+n. Stores are predicated; WMMA was not.
#pragma unroll
        for (int e = 0; e < 8; ++e) {
            const int row = R0h + e + 8 * hh;
            if (row < RROWS) {
                const int o = row * HS + c0h + n;
                lds[OFF_H + 0 * HSZ + o] = dacc[0][e];
                lds[OFF_H + 1 * HSZ + o] = dacc[1][e];
                lds[OFF_H + 2 * HSZ + o] = dacc[2][e];
                lds[OFF_H + 3 * HSZ + o] = dacc[3][e];
                lds[OFF_H + 4 * HSZ + o] = dacc[4][e];
            }
        }
    }
    __syncthreads();

    // ------------- Phase C: vertical pass on WMMA + SSIM -------------------
    // D(16x16 out) = A(weights 16x28 Toeplitz) x B(hconv data 28x16).
    for (int sub = wave; sub < 16; sub += 8) {  // uniform per wave
        const int R0 = (sub >> 2) * 16;  // out-row block base (== hX row base)
        const int c0 = (sub & 3) * 16;
        v8f acc[5];
#pragma unroll
        for (int m5 = 0; m5 < 5; ++m5)
            acc[m5] = (v8f){0.f, 0.f, 0.f, 0.f, 0.f, 0.f, 0.f, 0.f};
#pragma unroll
        for (int j = 0; j < 7; ++j) {
            const int sx = 4 * j + (hh ? 2 : 0);
            int r0 = R0 + sx;     if (r0 > RROWS - 1) r0 = RROWS - 1;
            int r1 = R0 + sx + 1; if (r1 > RROWS - 1) r1 = RROWS - 1;
            v2f av; av.x = wwx[j]; av.y = wwy[j];
#pragma unroll
            for (int m5 = 0; m5 < 5; ++m5) {
                v2f bv;
                bv.x = lds[OFF_H + m5 * HSZ + r0 * HS + c0 + n];
                bv.y = lds[OFF_H + m5 * HSZ + r1 * HS + c0 + n];
                acc[m5] = __builtin_amdgcn_wmma_f32_16x16x4_f32(
                    false, av, false, bv, (short)0, acc[m5], false, false);
            }
        }
        // Each lane holds 8 aligned outputs per map (D layout).
#pragma unroll
        for (int e = 0; e < 8; ++e) {
            const float mu1 = acc[0][e], mu2 = acc[1][e];
            const float xAA = acc[2][e], xBB = acc[3][e], xAB = acc[4][e];
            const float mu11 = mu1 * mu1, mu22 = mu2 * mu2, mu12 = mu1 * mu2;
            const float s11 = xAA - mu11, s22 = xBB - mu22, s12 = xAB - mu12;
            const float num = (2.f * mu12 + C1c) * (2.f * s12 + C2c);
            const float den = (mu11 + mu22 + C1c) * (s11 + s22 + C2c);
            lsum += num / den;
        }
    }
#else
    // ------------- VALU fallback (no WMMA builtin) -------------------------
    for (int task = tid; task < RROWS * 8; task += 256) {
        const int r  = task >> 3;
        const int c0 = (task & 7) * 8;
        float a[18], b[18], paa[18], pbb[18], pab[18];
#pragma unroll
        for (int u = 0; u < 18; ++u) {
            const float x = lds[r * RS + c0 + u];
            const float y = lds[OFF_RAWB + r * RS + c0 + u];
            a[u] = x; b[u] = y;
            paa[u] = x * x; pbb[u] = y * y; pab[u] = x * y;
        }
#pragma unroll
        for (int i2 = 0; i2 < 8; ++i2) {
            float s0 = 0.f, s1 = 0.f, s2 = 0.f, s3 = 0.f, s4 = 0.f;
#pragma unroll
            for (int t = 0; t < 11; ++t) {
                const float wt = KW[t];
                s0 += wt * a[i2 + t];
                s1 += wt * b[i2 + t];
                s2 += wt * paa[i2 + t];
                s3 += wt * pbb[i2 + t];
                s4 += wt * pab[i2 + t];
            }
            const int o = r * HS + c0 + i2;
            lds[OFF_H + 0 * HSZ + o] = s0;
            lds[OFF_H + 1 * HSZ + o] = s1;
            lds[OFF_H + 2 * HSZ + o] = s2;
            lds[OFF_H + 3 * HSZ + o] = s3;
            lds[OFF_H + 4 * HSZ + o] = s4;
        }
    }
    __syncthreads();
    for (int idx = tid; idx < TILE * TILE; idx += 256) {
        const int ro = idx >> 6;
        const int c  = idx & 63;
        float m1 = 0.f, m2 = 0.f, aa = 0.f, bb = 0.f, ab = 0.f;
#pragma unroll
        for (int t = 0; t < 11; ++t) {
            const float wt = KW[t];
            const int o = (ro + t) * HS + c;
            m1 += wt * lds[OFF_H + 0 * HSZ + o];
            m2 += wt * lds[OFF_H + 1 * HSZ + o];
            aa += wt * lds[OFF_H + 2 * HSZ + o];
            bb += wt * lds[OFF_H + 3 * HSZ + o];
            ab += wt * lds[OFF_H + 4 * HSZ + o];
        }
        const float mu11 = m1 * m1, mu22 = m2 * m2, mu12 = m1 * m2;
        const float s11 = aa - mu11, s22 = bb - mu22, s12 = ab - mu12;
        const float num = (2.f * mu12 + C1c) * (2.f * s12 + C2c);
        const float den = (mu11 + mu22 + C1c) * (s11 + s22 + C2c);
        lsum += num / den;
    }
#endif

    // ---------------- Phase D: block reduction -> ws[block] ----------------
#pragma unroll
    for (int off = 16; off > 0; off >>= 1) lsum += __shfl_xor(lsum, off, 32);
    if (lane == 0) lds[OFF_RED + wave] = lsum;
    __syncthreads();
    if (tid == 0) {
        float s = 0.f;
#pragma unroll
        for (int i2 = 0; i2 < 8; ++i2) s += lds[OFF_RED + i2];
        ws[blockIdx.x] = s;
    }
}

__global__ __launch_bounds__(256) void ssim_finalize_kernel(
    const float* __restrict__ ws, float* __restrict__ out) {
    __shared__ float sb[8];
    float s = 0.f;
    for (int i = threadIdx.x; i < NBLOCKS; i += 256) s += ws[i];
#pragma unroll
    for (int off = 16; off > 0; off >>= 1) s += __shfl_xor(s, off, 32);
    if ((threadIdx.x & 31) == 0) sb[threadIdx.x >> 5] = s;
    __syncthreads();
    if (threadIdx.x == 0) {
        float tot = 0.f;
#pragma unroll
        for (int i = 0; i < 8; ++i) tot += sb[i];
        out[0] = 1.0f - tot * (float)(1.0 / 25165824.0);
    }
}

extern "C" void kernel_launch(void* const* d_in, const int* in_sizes, int n_in,
                              void* d_out, int out_size, void* d_ws,
                              size_t ws_size, hipStream_t stream) {
    (void)in_sizes; (void)n_in; (void)out_size; (void)ws_size;
    const float* img1 = (const float*)d_in[0];
    const float* img2 = (const float*)d_in[1];
    float* out = (float*)d_out;
    float* ws  = (float*)d_ws;  // NBLOCKS floats of scratch

    ssim_fused_kernel<<<NBLOCKS, 256, LDS_FLOATS * sizeof(float), stream>>>(
        img1, img2, ws);
    ssim_finalize_kernel<<<1, 256, 0, stream>>>(ws, out);
}